// LSSFPN_12120397709810
// MI455X (gfx1250) — compile-verified
//
#include <hip/hip_runtime.h>
#include <cstdint>
#include <cstddef>

// ---------------- problem constants (from reference) ----------------
#define B_   2
#define N_   6
#define D_   112
#define H_   16
#define W_   44
#define C_   80
#define HW_        (H_ * W_)          // 704
#define PIX_PER_B  (N_ * HW_)         // 4224
#define PTS_PER_B  (N_ * D_ * HW_)    // 473088
#define NVOX       16384              // 128 x 128
#define OUT_ELEMS  (B_ * C_ * NVOX)   // 2,621,440

// ---------------- tiling ----------------
#define CT      4                 // channels per block (LDS bev tile = 16384*4*4B = 256 KB)
#define CTILES  (C_ / CT)         // 20
#define TPB     512               // 16 wave32 per block (D_ = 16 * 7)
#define NWAVES  (TPB / 32)        // 16

typedef unsigned int v4u __attribute__((ext_vector_type(4)));
typedef int          v8i __attribute__((ext_vector_type(8)));
typedef int          v4i __attribute__((ext_vector_type(4)));

// ---------------- kernel 0: zero the output (harness poisons d_out) -------------
__global__ void lss_zero(float* __restrict__ out, int n) {
    int i = blockIdx.x * blockDim.x + threadIdx.x;
    if (i < n) out[i] = 0.0f;
}

// ---------------- kernel 1: wave-per-pixel softmax over D ----------------
// logits/probs layout: [B, N, D, H, W]; one wave32 handles one (b,n,h,w) pixel.
__global__ void lss_softmax(const float* __restrict__ logits, float* __restrict__ probs) {
    const int wave = (blockIdx.x * blockDim.x + threadIdx.x) >> 5;
    const int lane = threadIdx.x & 31;
    if (wave >= B_ * PIX_PER_B) return;

    const int b  = wave / PIX_PER_B;
    const int r  = wave - b * PIX_PER_B;
    const int n  = r / HW_;
    const int hw = r - n * HW_;
    const size_t base = (size_t)b * PTS_PER_B + (size_t)n * (D_ * HW_) + (size_t)hw;

    float v[4];
    float mx = -3.4e38f;
#pragma unroll
    for (int i = 0; i < 4; ++i) {
        const int d = lane + 32 * i;
        v[i] = (d < D_) ? logits[base + (size_t)d * HW_] : -3.4e38f;
        mx = fmaxf(mx, v[i]);
    }
#pragma unroll
    for (int off = 16; off >= 1; off >>= 1)
        mx = fmaxf(mx, __shfl_xor(mx, off, 32));

    float e[4];
    float s = 0.0f;
#pragma unroll
    for (int i = 0; i < 4; ++i) {
        const int d = lane + 32 * i;
        e[i] = (d < D_) ? __expf(v[i] - mx) : 0.0f;
        s += e[i];
    }
#pragma unroll
    for (int off = 16; off >= 1; off >>= 1)
        s += __shfl_xor(s, off, 32);

    const float inv = 1.0f / s;
#pragma unroll
    for (int i = 0; i < 4; ++i) {
        const int d = lane + 32 * i;
        if (d < D_) probs[base + (size_t)d * HW_] = e[i] * inv;
    }
}

// ---------------- kernel 2: lift + splat with LDS-privatized BEV ----------------
// Block = (b, 4-channel tile c0, camera n): 2*20*6 = 240 blocks, one per WGP.
// LDS holds a private fp32 BEV tile [NVOX][CT] (256 KB) plus a TDM-staged context
// slab [CT][HW_] for this camera. Scatter uses ds_add_f32 LDS atomics; flush uses
// fp32 global atomics (N_ racing copies per (b,c) tile -> 15.7M global atomics
// total vs 75.7M without privatization).
__global__ void lss_splat(const float* __restrict__ probs,
                          const float* __restrict__ ctx,
                          const int*   __restrict__ geom,
                          float*       __restrict__ out) {
    extern __shared__ float lds[];           // [NVOX*CT] accum | [CT*HW_] ctx stage
    const int bid = blockIdx.x;
    const int b   = bid / (CTILES * N_);
    const int r   = bid - b * (CTILES * N_);
    const int ct  = r / N_;
    const int n   = r - ct * N_;
    const int c0  = ct * CT;

    for (int i = threadIdx.x; i < NVOX * CT; i += blockDim.x) lds[i] = 0.0f;
    float* ctxs = lds + NVOX * CT;

    // wave 0: TDM-stage context[b, n, c0:c0+CT, :] (CT x 704 f32, row stride 704)
    if (threadIdx.x < 32) {
        const unsigned long long ga = (unsigned long long)(uintptr_t)(ctx +
            (size_t)b * (N_ * C_ * HW_) + (size_t)n * (C_ * HW_) + (size_t)c0 * HW_);
        const unsigned int lds_off = (unsigned int)(NVOX * CT * sizeof(float));
        v4u g0;
        g0.x = 1u;                                                   // count=1 (valid D#)
        g0.y = lds_off;                                              // lds_addr (bytes)
        g0.z = (unsigned int)(ga & 0xFFFFFFFFull);                   // global_addr[31:0]
        g0.w = (unsigned int)((ga >> 32) & 0x01FFFFFFull)            // global_addr[56:32]
             | (2u << 30);                                           // type = 2 ("image")
        v8i g1;
        g1[0] = (int)(2u << 16);                 // data_size = 4 bytes
        g1[1] = (int)((unsigned)HW_ << 16);      // tensor_dim0[15:0] = 704
        g1[2] = (int)((unsigned)C_  << 16);      // tensor_dim0[31:16]=0 | tensor_dim1 = 80
        g1[3] = (int)((unsigned)HW_ << 16);      // tile_dim0 = 704
        g1[4] = (int)CT;                         // tile_dim1 = 4, tile_dim2 = 0
        g1[5] = (int)HW_;                        // tensor_dim0_stride = 704
        g1[6] = 0;                               // stride hi / dim1_stride (unused, 2D)
        g1[7] = 0;
        v4i g2 = {0, 0, 0, 0};                   // 2D tensor: groups 2/3 unused
        v4i g3 = {0, 0, 0, 0};
        v8i g4 = {0, 0, 0, 0, 0, 0, 0, 0};       // extra group (6-arg toolchain variant)
        __builtin_amdgcn_tensor_load_to_lds(g0, g1, g2, g3, g4, 0);
        __builtin_amdgcn_s_wait_tensorcnt(0);
    }
    __syncthreads();

    const int wv   = threadIdx.x >> 5;
    const int lane = threadIdx.x & 31;

    // scatter: waves stride depth bins (112 = 16*7), lanes stride contiguous W
    for (int d = wv; d < D_; d += NWAVES) {
        const size_t dbase = (size_t)b * PTS_PER_B + (size_t)n * (D_ * HW_) +
                             (size_t)d * HW_;
        for (int pp = lane; pp < HW_; pp += 32) {
            const size_t off = dbase + pp;
            // prefetch this wave's next depth slice
            __builtin_prefetch(probs + off + (size_t)NWAVES * HW_, 0, 1);
            __builtin_prefetch(geom  + off + (size_t)NWAVES * HW_, 0, 1);
            const float p = probs[off];
            const int   v = geom[off];
            float* dst = &lds[(unsigned)v * CT];
#pragma unroll
            for (int c = 0; c < CT; ++c)
                atomicAdd(dst + c, p * ctxs[c * HW_ + pp]);   // ds_add_f32
        }
    }
    __syncthreads();

    // flush private BEV tile: coalesced over voxels, fp32 global atomics
    for (int v = threadIdx.x; v < NVOX; v += blockDim.x) {
#pragma unroll
        for (int c = 0; c < CT; ++c)
            unsafeAtomicAdd(out + ((size_t)(b * C_ + c0 + c) * NVOX + v),
                            lds[v * CT + c]);
    }
}

// ---------------- host-side launch ----------------
extern "C" void kernel_launch(void* const* d_in, const int* in_sizes, int n_in,
                              void* d_out, int out_size, void* d_ws, size_t ws_size,
                              hipStream_t stream) {
    const float* depth_logits = (const float*)d_in[0];   // [B,N,D,H,W] f32
    const float* context      = (const float*)d_in[1];   // [B,N,C,H,W] f32
    const int*   geom_idx     = (const int*)  d_in[2];   // [B,N,D,H,W] i32
    float*       out          = (float*)d_out;           // [B,C,128,128] f32
    float*       probs        = (float*)d_ws;            // [B,N,D,H,W] f32 (~3.8 MB)

    // zero output (harness poisons it with 0xAA)
    lss_zero<<<(OUT_ELEMS + 1023) / 1024, 1024, 0, stream>>>(out, OUT_ELEMS);

    // softmax over depth: 8448 pixels, one wave32 per pixel
    const int nwaves = B_ * PIX_PER_B;                   // 8448
    lss_softmax<<<(nwaves * 32 + 255) / 256, 256, 0, stream>>>(depth_logits, probs);

    // lift + splat: 240 blocks (one per WGP), 512 threads, 256KB bev + 11KB ctx in LDS
    const size_t lds_bytes = (size_t)(NVOX * CT + CT * HW_) * sizeof(float); // 273,408 B
    lss_splat<<<B_ * CTILES * N_, TPB, lds_bytes, stream>>>(probs, context, geom_idx, out);
}